// MoELayer_69638599737867
// MI455X (gfx1250) — compile-verified
//
#include <hip/hip_runtime.h>
#include <math.h>

// ---------------- problem constants (match reference) ----------------
#define T_TOK 8192   // B*S
#define DM    1024   // d_model
#define DFF   4096   // d_ff
#define NE    8      // experts
#define CAP   1024   // int(1.0 * T / E)

typedef __attribute__((ext_vector_type(2))) float v2f;
typedef __attribute__((ext_vector_type(8))) float v8f;

// Exact-fp32 WMMA: D(16x16,f32) = A(16x4,f32) * B(4x16,f32) + C
__device__ __forceinline__ v8f wmma4(v2f a, v2f b, v8f c) {
  return __builtin_amdgcn_wmma_f32_16x16x4_f32(
      false, a, false, b, (short)0, c, false, false);
}

// ---- CDNA5 async memory->LDS (ISA 15.18.3 op 98, ASYNCcnt tracked) ----
__device__ __forceinline__ void async_lds_b128(const float* gptr, unsigned ldsoff) {
  asm volatile("global_load_async_to_lds_b128 %0, %1, off"
               :: "v"(ldsoff), "v"(gptr) : "memory");
}
__device__ __forceinline__ void wait_async_le3() {
  asm volatile("s_wait_asynccnt 0x3" ::: "memory");
}
__device__ __forceinline__ void wait_async_le0() {
  asm volatile("s_wait_asynccnt 0x0" ::: "memory");
}
__device__ __forceinline__ unsigned lds_off(const void* p) {
  // generic LDS pointer: low 32 bits are the LDS byte address (ISA 10.2)
  return (unsigned)(size_t)p;
}

// ---------------- 1) router: logits -> top2 -> weights, counts ----------------
__global__ void __launch_bounds__(256) moe_router(
    const float* __restrict__ x, const float* __restrict__ rw,
    int* __restrict__ tok_expert, float* __restrict__ tok_weight,
    int* __restrict__ counts) {
  __shared__ float srw[NE * DM];  // 32 KB: router weights resident in LDS
  const int tid = threadIdx.x;
  for (int i = tid; i < NE * DM; i += 256) srw[i] = rw[i];
  __syncthreads();

  const int lane = tid & 31;
  const int wave = tid >> 5;  // 8 waves -> 8 tokens per block
  const int t = blockIdx.x * 8 + wave;
  const float* xr = x + (size_t)t * DM;

  float acc[NE];
#pragma unroll
  for (int e = 0; e < NE; ++e) acc[e] = 0.f;
  for (int d = lane; d < DM; d += 32) {
    const float xv = xr[d];
#pragma unroll
    for (int e = 0; e < NE; ++e) acc[e] = fmaf(xv, srw[e * DM + d], acc[e]);
  }
#pragma unroll
  for (int e = 0; e < NE; ++e) {
#pragma unroll
    for (int off = 16; off >= 1; off >>= 1)
      acc[e] += __shfl_xor(acc[e], off, 32);
  }

  if (lane == 0) {
    int i0 = 0;
#pragma unroll
    for (int e = 1; e < NE; ++e)
      if (acc[e] > acc[i0]) i0 = e;  // strict > keeps first on ties
    int i1 = (i0 == 0) ? 1 : 0;
#pragma unroll
    for (int e = 0; e < NE; ++e)
      if (e != i0 && acc[e] > acc[i1]) i1 = e;
    // normalized top-2 softmax weights: p0/(p0+p1) = 1/(1+exp(l1-l0))
    const float w0 = 1.f / (1.f + expf(acc[i1] - acc[i0]));
    tok_expert[2 * t + 0] = i0;
    tok_expert[2 * t + 1] = i1;
    tok_weight[2 * t + 0] = w0;
    tok_weight[2 * t + 1] = 1.f - w0;
    atomicAdd(&counts[i0], 1);
    atomicAdd(&counts[i1], 1);
  }
}

// ------- 2) capacity-limited dispatch: stable prefix scan, one wave/expert -------
__global__ void moe_dispatch(const int* __restrict__ tok_expert,
                             int* __restrict__ tok_slot,
                             int* __restrict__ idx_ec) {
  const int e = blockIdx.x;
  const int lane = threadIdx.x;  // wave32
  int running = 0;
  for (int base = 0; base < T_TOK; base += 32) {
    const int t = base + lane;
    const int e0 = tok_expert[2 * t];
    const int e1 = tok_expert[2 * t + 1];
    const int k = (e0 == e) ? 0 : ((e1 == e) ? 1 : -1);
    const unsigned mask = (unsigned)__ballot(k >= 0);
    if (k >= 0) {
      const int pos = running + __popc(mask & ((1u << lane) - 1u));
      tok_slot[2 * t + k] = (pos < CAP) ? pos : -1;
      if (pos < CAP) idx_ec[e * CAP + pos] = t;
    }
    running += __popc(mask);
  }
  for (int p = running + lane; p < CAP; p += 32) idx_ec[e * CAP + p] = -1;
}

// ---------------- 3) load-balance loss ----------------
__global__ void moe_lbloss(const int* __restrict__ counts,
                           float* __restrict__ out_loss) {
  const int lane = threadIdx.x;
  float v = 0.f;
  if (lane < NE) {
    const float d = (float)counts[lane] - (float)(T_TOK * 2 / NE);  // ideal=2048
    v = d * d;
  }
#pragma unroll
  for (int off = 16; off >= 1; off >>= 1) v += __shfl_xor(v, off, 32);
  if (lane == 0) *out_loss = v / ((float)T_TOK * (float)T_TOK);
}

// ---------------- grouped GEMM: async double-buffered LDS + f32 WMMA ----------------
template <bool GELU>
__device__ __forceinline__ void store8(float* __restrict__ cb, long row,
                                       long col, long Nt, v8f c, float bias) {
#pragma unroll
  for (int r = 0; r < 8; ++r) {
    float v = c[r] + bias;
    if (GELU) v = 0.5f * v * (1.f + erff(v * 0.7071067811865475f));
    cb[(row + r) * Nt + col] = v;
  }
}

// Block tile 64(M) x 128(N); K staged 16-deep, double-buffered in LDS via
// global_load_async_to_lds_b128. 8 waves, each computing 32x32 with 2x2 WMMA.
// GATHER: A rows come from x via idx_ec (token gather; -1 rows are clamped to
// row 0 so every thread issues a uniform 3 async ops per tile -- garbage lands
// only in capacity slots that are never referenced by tok_slot).
template <bool GATHER, bool GELU>
__global__ void __launch_bounds__(256) moe_gemm(
    const float* __restrict__ A, const int* __restrict__ tok_ids,
    const float* __restrict__ Bw, const float* __restrict__ bias,
    float* __restrict__ Cout, int Kt, int Nt) {
  __shared__ float As[2][64][20];   // [m][k], 80B rows: 16B-aligned, conflict-free
  __shared__ float Bs[2][16][132];  // [k][n], 528B rows: 16B-aligned
  __shared__ int toks[64];

  const int tid = threadIdx.x;
  const int e = blockIdx.z;
  const int mt = blockIdx.y;  // 64-row tile of CAP
  const int nt = blockIdx.x;  // 128-col tile of Nt

  if (GATHER) {
    if (tid < 64) toks[tid] = tok_ids[e * CAP + mt * 64 + tid];
    __syncthreads();
  }

  const int lane = tid & 31;
  const int half = lane >> 4;  // 0: lanes 0-15, 1: lanes 16-31
  const int ln = lane & 15;
  const int wave = tid >> 5;
  const int Ma = (wave >> 2) * 32;  // wave M offset (0,32)
  const int Nb = (wave & 3) * 32;   // wave N offset (0,32,64,96)

  v8f c00 = {}, c01 = {}, c10 = {}, c11 = {};

  // cooperative async loaders (per thread: 1x A B128 + 2x B B128 per K-tile)
  const int lm = tid >> 2;        // A: row 0..63
  const int lk = (tid & 3) * 4;   // A: k quad
  const int bk = tid >> 4;        // B: k row 0..15
  const int bn = (tid & 15) * 8;  // B: col octet

  long arow;
  if (GATHER) {
    const int tk = toks[lm];
    arow = (tk < 0) ? 0 : tk;  // clamp: uniform issue count, confined garbage
  } else {
    arow = (long)e * CAP + (long)mt * 64 + lm;
  }
  const float* aptr = A + arow * (long)Kt + lk;
  const float* bptr = Bw + ((long)e * Kt + bk) * (long)Nt + (long)nt * 128 + bn;

  const unsigned asoff0 = lds_off(&As[0][lm][lk]);
  const unsigned asoff1 = lds_off(&As[1][lm][lk]);
  const unsigned bsoff0 = lds_off(&Bs[0][bk][bn]);
  const unsigned bsoff1 = lds_off(&Bs[1][bk][bn]);

  auto issue = [&](int buf, int k0) {
    async_lds_b128(aptr + k0, buf ? asoff1 : asoff0);
    const float* bp = bptr + (long)k0 * Nt;
    const unsigned bo = buf ? bsoff1 : bsoff0;
    async_lds_b128(bp, bo);
    async_lds_b128(bp + 4, bo + 16);
  };

  issue(0, 0);
  int cur = 0;
  for (int k0 = 0; k0 < Kt; k0 += 16) {
    const bool more = (k0 + 16) < Kt;
    if (more) issue(cur ^ 1, k0 + 16);  // prefetch next tile into other buffer
    if (more) wait_async_le3();         // oldest 3 ops (this tile) complete
    else      wait_async_le0();
    __syncthreads();

#pragma unroll
    for (int kk = 0; kk < 16; kk += 4) {
      const int kr = kk + 2 * half;  // ISA: lanes16-31 hold K=2,3 of the step
      v2f a0, a1, b0, b1;
      a0.x = As[cur][Ma + ln][kr];      a0.y = As[cur][Ma + ln][kr + 1];
      a1.x = As[cur][Ma + 16 + ln][kr]; a1.y = As[cur][Ma + 16 + ln][kr + 1];
      b0.x = Bs[cur][kr][Nb + ln];      b0.y = Bs[cur][kr + 1][Nb + ln];
      b1.x = Bs[cur][kr][Nb + 16 + ln]; b1.y = Bs[cur][kr + 1][Nb + 16 + ln];
      c00 = wmma4(a0, b0, c00);
      c01 = wmma4(a0, b1, c01);
      c10 = wmma4(a1, b0, c10);
      c11 = wmma4(a1, b1, c11);
    }
    __syncthreads();  // all waves done with buf `cur` before it is refilled
    cur ^= 1;
  }

  // epilogue: C layout VGPR r -> rows (base + r) / (base + 8 + r) split by half
  float* cb = Cout + (long)e * CAP * Nt;
  const long gm = (long)mt * 64 + Ma;
  const long gn = (long)nt * 128 + Nb;
  const float bias0 = bias[(long)e * Nt + gn + ln];
  const float bias1 = bias[(long)e * Nt + gn + 16 + ln];
  store8<GELU>(cb, gm + 8 * half,      gn + ln,      Nt, c00, bias0);
  store8<GELU>(cb, gm + 8 * half,      gn + 16 + ln, Nt, c01, bias1);
  store8<GELU>(cb, gm + 16 + 8 * half, gn + ln,      Nt, c10, bias0);
  store8<GELU>(cb, gm + 16 + 8 * half, gn + 16 + ln, Nt, c11, bias1);
}

// ---------------- combine: deterministic gather per token ----------------
__global__ void __launch_bounds__(256) moe_combine(
    const float* __restrict__ Y, const int* __restrict__ tok_expert,
    const int* __restrict__ tok_slot, const float* __restrict__ tok_weight,
    float* __restrict__ out) {
  const int t = blockIdx.x;
  const int e0 = tok_expert[2 * t], e1 = tok_expert[2 * t + 1];
  const int s0 = tok_slot[2 * t],   s1 = tok_slot[2 * t + 1];
  float w0 = tok_weight[2 * t], w1 = tok_weight[2 * t + 1];
  const float* y0 = Y + ((long)e0 * CAP + (s0 < 0 ? 0 : s0)) * DM;
  const float* y1 = Y + ((long)e1 * CAP + (s1 < 0 ? 0 : s1)) * DM;
  if (s0 < 0) w0 = 0.f;  // dropped by capacity
  if (s1 < 0) w1 = 0.f;
  for (int d = threadIdx.x; d < DM; d += 256)
    out[(long)t * DM + d] = w0 * y0[d] + w1 * y1[d];
}

// ---------------- host launcher ----------------
extern "C" void kernel_launch(void* const* d_in, const int* in_sizes, int n_in,
                              void* d_out, int out_size, void* d_ws,
                              size_t ws_size, hipStream_t stream) {
  (void)in_sizes; (void)n_in; (void)out_size; (void)ws_size;
  const float* x  = (const float*)d_in[0];
  const float* rw = (const float*)d_in[1];
  const float* w1 = (const float*)d_in[2];
  const float* b1 = (const float*)d_in[3];
  const float* w2 = (const float*)d_in[4];
  const float* b2 = (const float*)d_in[5];
  float* out = (float*)d_out;  // T*DM outputs + 1 lb_loss scalar

  char* ws = (char*)d_ws;
  size_t off = 0;
  auto wsalloc = [&](size_t nbytes) -> void* {
    off = (off + 255) & ~(size_t)255;
    void* p = ws + off;
    off += nbytes;
    return p;
  };
  int*   counts     = (int*)  wsalloc(NE * sizeof(int));
  int*   tok_expert = (int*)  wsalloc((size_t)T_TOK * 2 * sizeof(int));
  int*   tok_slot   = (int*)  wsalloc((size_t)T_TOK * 2 * sizeof(int));
  float* tok_weight = (float*)wsalloc((size_t)T_TOK * 2 * sizeof(float));
  int*   idx_ec     = (int*)  wsalloc((size_t)NE * CAP * sizeof(int));
  float* H          = (float*)wsalloc((size_t)NE * CAP * DFF * sizeof(float)); // 128 MB, L2-resident
  float* Y          = (float*)wsalloc((size_t)NE * CAP * DM  * sizeof(float)); // 32 MB

  hipMemsetAsync(counts, 0, NE * sizeof(int), stream);
  moe_router<<<T_TOK / 8, 256, 0, stream>>>(x, rw, tok_expert, tok_weight, counts);
  moe_dispatch<<<NE, 32, 0, stream>>>(tok_expert, tok_slot, idx_ec);
  moe_lbloss<<<1, 32, 0, stream>>>(counts, out + (size_t)T_TOK * DM);
  moe_gemm<true, true><<<dim3(DFF / 128, CAP / 64, NE), 256, 0, stream>>>(
      x, idx_ec, w1, b1, H, DM, DFF);
  moe_gemm<false, false><<<dim3(DM / 128, CAP / 64, NE), 256, 0, stream>>>(
      H, nullptr, w2, b2, Y, DFF, DM);
  moe_combine<<<T_TOK, 256, 0, stream>>>(Y, tok_expert, tok_slot, tok_weight, out);
}